// TotalVariationLoss_45475113730210
// MI455X (gfx1250) — compile-verified
//
#include <hip/hip_runtime.h>

typedef __attribute__((ext_vector_type(2))) float v2f;
typedef __attribute__((ext_vector_type(8))) float v8f;

#define NB    4
#define NPTS  8192
#define KNN   8
#define CHUNK 1024   // database points staged in LDS per pipeline stage (16 KB/buffer)

__global__ void tv_zero_kernel(float* __restrict__ out) {
    if (threadIdx.x == 0) out[0] = 0.0f;
}

// Pack augmented coordinates once:
//   arrA[i] = ( x,  y,  z, ||p||^2 )   (WMMA A-matrix rows, database points)
//   arrB[i] = (-2x,-2y,-2z, 1      )   (WMMA B-matrix cols, query points)
__global__ __launch_bounds__(256) void tv_pack_kernel(const float* __restrict__ pts,
                                                      float4* __restrict__ arrA,
                                                      float4* __restrict__ arrB) {
    const int i = blockIdx.x * blockDim.x + threadIdx.x;   // 0 .. NB*NPTS-1
    const float x = pts[3 * i + 0];
    const float y = pts[3 * i + 1];
    const float z = pts[3 * i + 2];
    const float sq = x * x + y * y + z * z;
    arrA[i] = make_float4(x, y, z, sq);
    arrB[i] = make_float4(-2.0f * x, -2.0f * y, -2.0f * z, 1.0f);
}

__device__ __forceinline__ void top8_insert(float (&t)[KNN], float d) {
    if (d < t[KNN - 1]) {
        t[KNN - 1] = d;
#pragma unroll
        for (int j = KNN - 1; j >= 1; --j) {   // single bubble pass
            const float lo = fminf(t[j - 1], t[j]);
            const float hv = fmaxf(t[j - 1], t[j]);
            t[j - 1] = lo;
            t[j]     = hv;
        }
    }
}

__device__ __forceinline__ void tv_finish(float (&t)[KNN], float* out, int lane) {
    // merge row-halves: lane c <-> lane c+16 (snapshot partner list first)
    float o[KNN];
#pragma unroll
    for (int i = 0; i < KNN; ++i) o[i] = __shfl_xor(t[i], 16, 32);
#pragma unroll
    for (int i = 0; i < KNN; ++i) top8_insert(t, o[i]);

    float s = 0.0f;
#pragma unroll
    for (int i = 0; i < KNN; ++i) s += t[i];
#pragma unroll
    for (int off = 16; off >= 1; off >>= 1) s += __shfl_xor(s, off, 32);

    if (lane == 0) {
        const float scale = 0.5f / ((float)NB * (float)NPTS * (float)KNN);
        atomicAdd(out, s * scale);
    }
}

// Fast path: 8 waves/block share one batch. Double-buffered LDS pipeline:
// async global->LDS DMA (ASYNCcnt) for chunk k+1 overlaps WMMA + top-8 work on
// chunk k; A-fragments software-pipelined one tile ahead through ds_load_b64.
__global__ __launch_bounds__(256) void tv_knn_wmma_lds_kernel(const float4* __restrict__ arrA,
                                                              const float4* __restrict__ arrB,
                                                              float* __restrict__ out) {
    __shared__ float4 sA[2][CHUNK];              // 2 x 16 KB staged database tiles

    const int lane   = threadIdx.x & 31;
    const int wid    = threadIdx.x >> 5;         // 0..7 (8 waves per block)
    const int b      = blockIdx.x >> 6;          // 64 blocks per batch
    const int qgroup = blockIdx.x & 63;
    const int qtile  = qgroup * 8 + wid;         // 0..511
    const int qbase  = qtile * 16;
    const int nl     = lane & 15;                // column n / A-row within tile
    const int hi     = lane >> 4;                // K-half for frags, M-half for C rows

    const float4* __restrict__ A  = arrA + (size_t)b * NPTS;
    const float4* __restrict__ Bq = arrB + (size_t)b * NPTS;

    // query (B-matrix) fragment + query norm, constant over the scan
    const int q = qbase + nl;
    const float4 qv = Bq[q];                     // (-2x,-2y,-2z,1)
    v2f bfrag;
    bfrag[0] = hi ? qv.z : qv.x;
    bfrag[1] = hi ? qv.w : qv.y;
    const float sqn = A[q].w;

    float t[KNN];
#pragma unroll
    for (int i = 0; i < KNN; ++i) t[i] = __builtin_inff();

    // issue the per-thread share (4 x b128) of one chunk's async DMA
    auto stage = [&](int base, int buf) {
#pragma unroll
        for (int i = 0; i < CHUNK / 256; ++i) {
            const int idx = (int)threadIdx.x + i * 256;
            const unsigned laddr = (unsigned)(uintptr_t)(&sA[buf][idx]);
            const unsigned long long ga =
                (unsigned long long)(const void*)(A + base + idx);
            asm volatile("global_load_async_to_lds_b128 %0, %1, off"
                         :: "v"(laddr), "v"(ga)
                         : "memory");
        }
    };

    // prologue: fill buffer 0
    stage(0, 0);
    asm volatile("s_wait_asynccnt 0" ::: "memory");
    __syncthreads();

    for (int c = 0; c < NPTS / CHUNK; ++c) {
        const int base = c * CHUNK;
        const int buf  = c & 1;

        // overlap: DMA chunk c+1 into the other buffer while computing chunk c
        if (c + 1 < NPTS / CHUNK) stage(base + CHUNK, buf ^ 1);

        const float4* __restrict__ sbuf = &sA[buf][0];

        // software-pipelined A-fragment: load tile mt+16 before consuming tile mt
        v2f af = *(const v2f*)((const char*)&sbuf[nl] + hi * 8);   // ds_load_b64
#pragma unroll 2
        for (int mt = 0; mt < CHUNK; mt += 16) {
            const v2f cur = af;
            if (mt + 16 < CHUNK)
                af = *(const v2f*)((const char*)&sbuf[mt + 16 + nl] + hi * 8);

            v8f cacc = {};
            cacc = __builtin_amdgcn_wmma_f32_16x16x4_f32(
                       false, cur, false, bfrag, (short)0, cacc, false, false);

#pragma unroll
            for (int r = 0; r < 8; ++r) {
                const int mrow = base + mt + r + hi * 8;
                float d2 = cacc[r] + sqn;
                if (mrow == q) d2 = __builtin_inff();   // exclude self
                top8_insert(t, d2);
            }
        }

        // chunk c+1 resident + all waves done reading buf before it is reused
        asm volatile("s_wait_asynccnt 0" ::: "memory");
        __syncthreads();
    }

    tv_finish(t, out, lane);
}

// Fallback (no workspace needed): builds fragments from raw points on the fly.
__global__ __launch_bounds__(256) void tv_knn_wmma_direct_kernel(const float* __restrict__ pts,
                                                                 float* __restrict__ out) {
    const int lane  = threadIdx.x & 31;
    const int wave  = (int)((blockIdx.x * blockDim.x + threadIdx.x) >> 5);
    const int b     = wave >> 9;
    const int qtile = wave & 511;
    const int qbase = qtile * 16;
    const int nl    = lane & 15;
    const int hi    = lane >> 4;

    const float* __restrict__ P = pts + (size_t)b * (size_t)NPTS * 3u;

    const int q = qbase + nl;
    const float qx = P[3 * q + 0];
    const float qy = P[3 * q + 1];
    const float qz = P[3 * q + 2];
    const float sqn = qx * qx + qy * qy + qz * qz;
    v2f bfrag;
    if (hi) { bfrag[0] = -2.0f * qz; bfrag[1] = 1.0f;       }
    else    { bfrag[0] = -2.0f * qx; bfrag[1] = -2.0f * qy; }

    float t[KNN];
#pragma unroll
    for (int i = 0; i < KNN; ++i) t[i] = __builtin_inff();

    for (int mt = 0; mt < NPTS; mt += 16) {
        const int m = mt + nl;
        const float px = P[3 * m + 0];
        const float py = P[3 * m + 1];
        const float pz = P[3 * m + 2];
        v2f afrag;
        if (hi) { afrag[0] = pz; afrag[1] = px * px + py * py + pz * pz; }
        else    { afrag[0] = px; afrag[1] = py; }

        v8f c = {};
        c = __builtin_amdgcn_wmma_f32_16x16x4_f32(
                false, afrag, false, bfrag, (short)0, c, false, false);

#pragma unroll
        for (int r = 0; r < 8; ++r) {
            const int mrow = mt + r + hi * 8;
            float d2 = c[r] + sqn;
            if (mrow == q) d2 = __builtin_inff();
            top8_insert(t, d2);
        }
    }

    tv_finish(t, out, lane);
}

extern "C" void kernel_launch(void* const* d_in, const int* in_sizes, int n_in,
                              void* d_out, int out_size, void* d_ws, size_t ws_size,
                              hipStream_t stream) {
    (void)in_sizes; (void)n_in; (void)out_size;
    const float* pts = (const float*)d_in[0];
    float* out = (float*)d_out;

    tv_zero_kernel<<<1, 32, 0, stream>>>(out);

    const size_t need = (size_t)2 * NB * NPTS * sizeof(float4);  // 1 MB
    if (ws_size >= need) {
        float4* arrA = (float4*)d_ws;
        float4* arrB = arrA + (size_t)NB * NPTS;
        tv_pack_kernel<<<(NB * NPTS) / 256, 256, 0, stream>>>(pts, arrA, arrB);
        // NB*512 query tiles, 8 per block -> 256 blocks
        tv_knn_wmma_lds_kernel<<<256, 256, 0, stream>>>(arrA, arrB, out);
    } else {
        tv_knn_wmma_direct_kernel<<<256, 256, 0, stream>>>(pts, out);
    }
}